// multi_output_model_67654324847109
// MI455X (gfx1250) — compile-verified
//
#include <hip/hip_runtime.h>

// ---------------------------------------------------------------------------
// Problem constants (from reference): N=65536, D=1024, DF=32, ND=2, NC=5
// ---------------------------------------------------------------------------
#define NROWS 65536
#define DFEAT 1024
#define DF    32
#define BK    64          // K-chunk staged in LDS per iteration
#define LDP   68          // padded LDS row stride (floats): conflict-free ds_load_b64
#define JPAD  48          // 32 (x1_w rows) + 16 (y2_w rows, 5 real + 11 zero)
#define JREAL 37          // rows that actually need restaging each chunk

typedef float v2f __attribute__((ext_vector_type(2)));
typedef float v8f __attribute__((ext_vector_type(8)));
typedef int   v4i __attribute__((ext_vector_type(4)));

#ifndef __has_builtin
#define __has_builtin(x) 0
#endif
#if defined(__AMDGCN__) && __has_builtin(__builtin_amdgcn_global_load_async_to_lds_b128)
#define USE_ASYNC 1
#define ASG __attribute__((address_space(1)))
#define ASL __attribute__((address_space(3)))
// direct global -> LDS copy, tracked by ASYNCcnt (no VGPR round trip).
// builtin signature (from clang diagnostic): (v4i AS1*, v4i AS3*, imm offset, imm cpol)
#define ASYNC_CP128(gp, lp)                                                    \
    __builtin_amdgcn_global_load_async_to_lds_b128((ASG v4i*)(void*)(gp),      \
                                                   (ASL v4i*)(void*)(lp), 0, 0)
#else
#define USE_ASYNC 0
#endif

// 16-lane butterfly sum (rows live across a 16-lane half of the wave32)
#define R16(z) { z += __shfl_xor(z, 1, 16); z += __shfl_xor(z, 2, 16); \
                 z += __shfl_xor(z, 4, 16); z += __shfl_xor(z, 8, 16); }

// ---------------------------------------------------------------------------
// K1: fused pass over x. Per wave: one 16-row tile.
//   C0,C1 = x_tile @ x1_w^T   (j = 0..31, two 16x16 WMMA tiles)
//   C2    = x_tile @ y2pad^T  (jj = 0..15, rows >=5 zero => y_disease in 0..4)
// Epilogue per row: relu+bias -> f; y_dataset (K=32); y_padded (K=32).
// Writes: out_d[n*5+jj] = y_disease, ws_p[n*5+jj] = y_padded, out_yd[n*2+c].
// ---------------------------------------------------------------------------
__global__ __launch_bounds__(256)
void fused_main(const float* __restrict__ x,
                const float* __restrict__ x1w,
                const float* __restrict__ x1b,
                const float* __restrict__ y1w,
                const float* __restrict__ y1b,
                const float* __restrict__ y2w,
                const float* __restrict__ y2b,
                float* __restrict__ out_d,
                float* __restrict__ out_yd,
                float* __restrict__ ws_p)
{
    __shared__ float sW[JPAD * LDP];     // 48 x 64 weights (padded rows)
    __shared__ float sX[128 * LDP];      // 128 x 64 x-tile (8 waves x 16 rows)

    const int t    = threadIdx.x;
    const int lane = t & 31;
    const int wv   = t >> 5;             // wave id 0..7
    const int jr   = lane & 15;          // column-within-tile / row-within-tile
    const int half = lane >> 4;          // K-half selector for WMMA fragments
    const int rowBase = blockIdx.x * 128;

    // zero-fill the 11 padding weight rows (j = 37..47) once; they are
    // k-invariant so they never need restaging inside the k-loop.
    for (int idx = t; idx < (JPAD - JREAL) * 16; idx += 256) {
        int j  = JREAL + (idx >> 4);
        int kk = (idx & 15) * 4;
        *(float4*)(&sW[j * LDP + kk]) = make_float4(0.f, 0.f, 0.f, 0.f);
    }

    v8f c0 = {}; v8f c1 = {}; v8f c2 = {};

    for (int k0 = 0; k0 < DFEAT; k0 += BK) {
        // ---- stage weights: 37 real rows x 64 floats (592 x b128) ----
        #pragma unroll
        for (int p = 0; p < 3; ++p) {
            int idx = p * 256 + t;
            if (idx < JREAL * 16) {
                int j  = idx >> 4;
                int kk = (idx & 15) * 4;
                const float* src = (j < DF) ? (x1w + j * DFEAT + k0 + kk)
                                            : (y2w + (j - DF) * DFEAT + k0 + kk);
#if USE_ASYNC
                ASYNC_CP128(src, &sW[j * LDP + kk]);
#else
                *(float4*)(&sW[j * LDP + kk]) = *(const float4*)src;
#endif
            }
        }
        // ---- stage x: 128 rows x 64 floats (2048 x b128, coalesced) ----
        #pragma unroll
        for (int p = 0; p < 8; ++p) {
            int idx = p * 256 + t;
            int rl  = idx >> 4;          // 0..127
            int kk  = (idx & 15) * 4;
            const float* src = x + (size_t)(rowBase + rl) * DFEAT + k0 + kk;
#if USE_ASYNC
            ASYNC_CP128(src, &sX[rl * LDP + kk]);
#else
            *(float4*)(&sX[rl * LDP + kk]) = *(const float4*)src;
#endif
        }
#if USE_ASYNC
#if __has_builtin(__builtin_amdgcn_s_wait_asynccnt)
        __builtin_amdgcn_s_wait_asynccnt(0);
#else
        asm volatile("s_wait_asynccnt 0x0" ::: "memory");
#endif
#endif
        __syncthreads();

        // ---- WMMA fragment base pointers (ISA 7.12.2 f32 layouts) ----
        // A 16x4: lane holds A[M = jr][K = 2*half + {0,1}]
        // B 4x16: lane holds B[K = 2*half + {0,1}][N = jr]
        const float* xrow = &sX[(wv * 16 + jr) * LDP + 2 * half];
        const float* wr0  = &sW[(jr)      * LDP + 2 * half];
        const float* wr1  = &sW[(16 + jr) * LDP + 2 * half];
        const float* wr2  = &sW[(32 + jr) * LDP + 2 * half];

        #pragma unroll
        for (int kk = 0; kk < BK; kk += 4) {
            v2f a  = *(const v2f*)(xrow + kk);
            v2f b0 = *(const v2f*)(wr0 + kk);
            v2f b1 = *(const v2f*)(wr1 + kk);
            v2f b2 = *(const v2f*)(wr2 + kk);
            c0 = __builtin_amdgcn_wmma_f32_16x16x4_f32(false, a, false, b0,
                                                       (short)0, c0, false, false);
            c1 = __builtin_amdgcn_wmma_f32_16x16x4_f32(false, a, false, b1,
                                                       (short)0, c1, false, false);
            c2 = __builtin_amdgcn_wmma_f32_16x16x4_f32(false, a, false, b2,
                                                       (short)0, c2, false, false);
        }
        __syncthreads();
    }

    // ---- epilogue: per-lane small parameters ----
    float xb_lo = x1b[jr];
    float xb_hi = x1b[16 + jr];
    float w1a0 = y1w[jr],      w1b0 = y1w[16 + jr];   // y1_w[0][j], j=jr / 16+jr
    float w1a1 = y1w[32 + jr], w1b1 = y1w[48 + jr];   // y1_w[1][j]
    float yb0 = y1b[0], yb1 = y1b[1];
    const float* y2t = y2w + (DFEAT - DF);            // last 32 cols of y2_w
    float w2a0 = y2t[0 * DFEAT + jr], w2b0 = y2t[0 * DFEAT + 16 + jr];
    float w2a1 = y2t[1 * DFEAT + jr], w2b1 = y2t[1 * DFEAT + 16 + jr];
    float w2a2 = y2t[2 * DFEAT + jr], w2b2 = y2t[2 * DFEAT + 16 + jr];
    float w2a3 = y2t[3 * DFEAT + jr], w2b3 = y2t[3 * DFEAT + 16 + jr];
    float w2a4 = y2t[4 * DFEAT + jr], w2b4 = y2t[4 * DFEAT + 16 + jr];
    float b20 = y2b[0], b21 = y2b[1], b22 = y2b[2], b23 = y2b[3], b24 = y2b[4];
    float myb2 = (jr < 5) ? y2b[jr] : 0.f;

    // D layout: VGPR v, lanes 0-15 -> row M=v, lanes 16-31 -> row M=v+8, col N=jr
    #pragma unroll
    for (int v = 0; v < 8; ++v) {
        float f0 = fmaxf(c0[v] + xb_lo, 0.f);   // f[n][jr]
        float f1 = fmaxf(c1[v] + xb_hi, 0.f);   // f[n][16+jr]
        float dv = c2[v] + myb2;                // y_disease[n][jr] (jr<5), else 0

        float t0 = f0 * w1a0 + f1 * w1b0;       // y_dataset partials
        float t1 = f0 * w1a1 + f1 * w1b1;
        float p0 = f0 * w2a0 + f1 * w2b0;       // y_padded partials
        float p1 = f0 * w2a1 + f1 * w2b1;
        float p2 = f0 * w2a2 + f1 * w2b2;
        float p3 = f0 * w2a3 + f1 * w2b3;
        float p4 = f0 * w2a4 + f1 * w2b4;
        R16(t0) R16(t1) R16(p0) R16(p1) R16(p2) R16(p3) R16(p4)
        p0 += b20; p1 += b21; p2 += b22; p3 += b23; p4 += b24;

        int n = rowBase + (wv << 4) + v + (half << 3);
        if (jr < 2) out_yd[n * 2 + jr] = (jr == 0) ? (t0 + yb0) : (t1 + yb1);
        float pv = (jr == 0) ? p0 : (jr == 1) ? p1 : (jr == 2) ? p2
                 : (jr == 3) ? p3 : p4;
        if (jr < 5) {
            out_d[n * 5 + jr] = dv;             // y_disease (rewritten by K4)
            ws_p [n * 5 + jr] = pv;             // y_padded
        }
    }
}

// ---------------------------------------------------------------------------
// K2: per-block partial sums of gram = p^T p (25) and ptd = p^T d (25).
// One row per thread; fixed butterfly tree -> deterministic (no fp atomics).
// ---------------------------------------------------------------------------
__global__ __launch_bounds__(256)
void gram_partial(const float* __restrict__ ws_p,
                  const float* __restrict__ dvals,
                  float* __restrict__ partial)
{
    int n = blockIdx.x * 256 + threadIdx.x;
    float p[5], d[5];
    #pragma unroll
    for (int i = 0; i < 5; ++i) { p[i] = ws_p[n * 5 + i]; d[i] = dvals[n * 5 + i]; }
    float acc[50];
    #pragma unroll
    for (int a = 0; a < 5; ++a)
        #pragma unroll
        for (int b = 0; b < 5; ++b) {
            acc[a * 5 + b]      = p[a] * p[b];
            acc[25 + a * 5 + b] = p[a] * d[b];
        }
    #pragma unroll
    for (int e = 0; e < 50; ++e) {
        acc[e] += __shfl_xor(acc[e], 16, 32);
        acc[e] += __shfl_xor(acc[e],  8, 32);
        acc[e] += __shfl_xor(acc[e],  4, 32);
        acc[e] += __shfl_xor(acc[e],  2, 32);
        acc[e] += __shfl_xor(acc[e],  1, 32);
    }
    __shared__ float sacc[8][50];
    int lane = threadIdx.x & 31, w = threadIdx.x >> 5;
    if (lane == 0) {
        #pragma unroll
        for (int e = 0; e < 50; ++e) sacc[w][e] = acc[e];
    }
    __syncthreads();
    if (threadIdx.x < 50) {
        float s = 0.f;
        #pragma unroll
        for (int w2 = 0; w2 < 8; ++w2) s += sacc[w2][threadIdx.x];
        partial[blockIdx.x * 50 + threadIdx.x] = s;
    }
}

// ---------------------------------------------------------------------------
// K3: sum 256 partials in fixed order, invert SPD 5x5 gram (Gauss-Jordan),
// M = inv(gram) @ ptd  -> ws_M (25 floats).
// ---------------------------------------------------------------------------
__global__ void solve_small(const float* __restrict__ partial,
                            float* __restrict__ Mout)
{
    __shared__ float g[25];
    __shared__ float tt[25];
    int e = threadIdx.x;
    if (e < 50) {
        float s = 0.f;
        for (int b = 0; b < 256; ++b) s += partial[b * 50 + e];
        if (e < 25) g[e] = s; else tt[e - 25] = s;
    }
    __syncthreads();
    if (threadIdx.x == 0) {
        float A[5][10];
        for (int r = 0; r < 5; ++r)
            for (int c = 0; c < 5; ++c) {
                A[r][c]     = g[r * 5 + c];
                A[r][5 + c] = (r == c) ? 1.f : 0.f;
            }
        for (int col = 0; col < 5; ++col) {
            float ip = 1.f / A[col][col];
            for (int c = 0; c < 10; ++c) A[col][c] *= ip;
            for (int r = 0; r < 5; ++r) {
                if (r == col) continue;
                float f = A[r][col];
                for (int c = 0; c < 10; ++c) A[r][c] -= f * A[col][c];
            }
        }
        for (int a = 0; a < 5; ++a)
            for (int c = 0; c < 5; ++c) {
                float s = 0.f;
                for (int b = 0; b < 5; ++b) s += A[a][5 + b] * tt[b * 5 + c];
                Mout[a * 5 + c] = s;
            }
    }
}

// ---------------------------------------------------------------------------
// K4: y_hex[n] = y_disease[n] - y_padded[n] @ M   (in place over out_d)
// ---------------------------------------------------------------------------
__global__ __launch_bounds__(256)
void finalize(const float* __restrict__ ws_p,
              const float* __restrict__ Mg,
              float* __restrict__ out_d)
{
    __shared__ float M[25];
    if (threadIdx.x < 25) M[threadIdx.x] = Mg[threadIdx.x];
    __syncthreads();
    int n = blockIdx.x * 256 + threadIdx.x;
    float p[5], d[5];
    #pragma unroll
    for (int i = 0; i < 5; ++i) { p[i] = ws_p[n * 5 + i]; d[i] = out_d[n * 5 + i]; }
    #pragma unroll
    for (int c = 0; c < 5; ++c) {
        float s = d[c];
        #pragma unroll
        for (int a = 0; a < 5; ++a) s -= p[a] * M[a * 5 + c];
        out_d[n * 5 + c] = s;
    }
}

// ---------------------------------------------------------------------------
extern "C" void kernel_launch(void* const* d_in, const int* in_sizes, int n_in,
                              void* d_out, int out_size, void* d_ws, size_t ws_size,
                              hipStream_t stream)
{
    const float* x   = (const float*)d_in[0];
    const float* x1w = (const float*)d_in[1];
    const float* x1b = (const float*)d_in[2];
    const float* y1w = (const float*)d_in[3];
    const float* y1b = (const float*)d_in[4];
    const float* y2w = (const float*)d_in[5];
    const float* y2b = (const float*)d_in[6];

    float* out_d  = (float*)d_out;                 // [N,5]: y_disease then y_hex
    float* out_yd = out_d + (size_t)NROWS * 5;     // [N,2]: y_dataset
    float* ws_p   = (float*)d_ws;                  // [N,5]: y_padded
    float* ws_par = ws_p + (size_t)NROWS * 5;      // [256*50] partials
    float* ws_M   = ws_par + 256 * 50;             // [25]

    fused_main  <<<NROWS / 128, 256, 0, stream>>>(x, x1w, x1b, y1w, y1b, y2w, y2b,
                                                  out_d, out_yd, ws_p);
    gram_partial<<<256, 256, 0, stream>>>(ws_p, out_d, ws_par);
    solve_small <<<1, 64, 0, stream>>>(ws_par, ws_M);
    finalize    <<<256, 256, 0, stream>>>(ws_p, ws_M, out_d);
}